// HIPABlock_56667798504116
// MI455X (gfx1250) — compile-verified
//
#include <hip/hip_runtime.h>

typedef __attribute__((ext_vector_type(2))) float v2f;
typedef __attribute__((ext_vector_type(4))) float v4f;
typedef __attribute__((ext_vector_type(8))) float v8f;

// ---- problem constants --------------------------------------------------
static constexpr int B = 8, C = 128, H = 512, W = 512, COUT = 128;
static constexpr int NTOK = 408;                       // total kept tokens / batch
static constexpr int NS[6]   = {1, 4, 16, 64, 256, 1024};
static constexpr int KS[6]   = {1, 1, 4, 19, 76, 307};
static constexpr int TOFF[7] = {0, 1, 2, 6, 25, 101, 408};

// ---- workspace layout (float offsets, compile-time) ---------------------
// pyramids stored [B][C][N_l], level 5 first
static constexpr long long PYR_OFF[6] = {
    1396736LL,  // l0 (1 cell)
    1392640LL,  // l1
    1376256LL,  // l2
    1310720LL,  // l3
    1048576LL,  // l4
    0LL         // l5 (B*C*1024)
};
static constexpr long long NORM_OFF[6] = {
    1408672LL, 1408640LL, 1408512LL, 1408000LL, 1405952LL, 1397760LL
};
static constexpr long long IDX_OFF    = 1408680LL;          // int region, B*408 ints
static constexpr long long NORMED_OFF = 1411944LL;          // B*408*128 floats

// ---- K1: 512x512 -> 32x32 max pool, one block per (b,c,gy) --------------
// x is a 1 GiB single-use stream (>> 192 MB L2): use non-temporal loads so
// the streaming pass does not evict the L2-resident pyramid/weight data.
__global__ void pool512_kernel(const float* __restrict__ x, float* __restrict__ pyr5) {
    int blk = blockIdx.x;
    int gy  = blk & 31;
    int bc  = blk >> 5;                      // b*C + c
    const v4f* src = (const v4f*)(x + (size_t)bc * (H * W) + (size_t)gy * 16 * W);
    int t  = threadIdx.x;
    int c4 = t & 127;                        // float4 column within row
    int r0 = t >> 7;                         // 0 or 1
    float m = -3.402823466e38f;
    #pragma unroll
    for (int j = 0; j < 8; ++j) {
        v4f v = __builtin_nontemporal_load(&src[(r0 + 2 * j) * 128 + c4]); // b128 NT, coalesced
        m = fmaxf(m, fmaxf(fmaxf(v[0], v[1]), fmaxf(v[2], v[3])));
    }
    __shared__ float sm[256];
    sm[t] = m;
    __syncthreads();
    if (t < 32) {
        float mm = -3.402823466e38f;
        #pragma unroll
        for (int i = 0; i < 4; ++i)
            mm = fmaxf(mm, fmaxf(sm[t * 4 + i], sm[128 + t * 4 + i]));
        pyr5[(size_t)bc * 1024 + gy * 32 + t] = mm;
    }
}

// ---- K2: 2g x 2g -> g x g max (hierarchy) -------------------------------
__global__ void down_kernel(const float* __restrict__ src, float* __restrict__ dst,
                            int g, int N) {
    int idx = blockIdx.x * blockDim.x + threadIdx.x;
    if (idx >= B * C * N) return;
    int cell = idx % N, bc = idx / N;
    int y = cell / g, xx = cell % g;
    int G = g * 2;
    const float* s = src + (size_t)bc * 4 * N + (size_t)(2 * y) * G + 2 * xx;
    dst[(size_t)bc * N + cell] = fmaxf(fmaxf(s[0], s[1]), fmaxf(s[G], s[G + 1]));
}

// ---- K3: channel L2 norm per cell ---------------------------------------
__global__ void norm_kernel(const float* __restrict__ pyr, float* __restrict__ nrm, int N) {
    int idx = blockIdx.x * blockDim.x + threadIdx.x;
    if (idx >= B * N) return;
    int cell = idx % N, b = idx / N;
    const float* p = pyr + (size_t)b * C * N + cell;
    float s = 0.f;
    #pragma unroll 8
    for (int c = 0; c < C; ++c) { float v = p[(size_t)c * N]; s = fmaf(v, v, s); }
    nrm[idx] = sqrtf(s);
}

// ---- K4: importance + exact top_k order via rank counting ---------------
__global__ void topk_kernel(const float* __restrict__ normCur, const float* __restrict__ normPar,
                            int N, int g, int k, int* __restrict__ idxOut,
                            float* __restrict__ coordsOut, int offTok) {
    int b = blockIdx.x;
    int i = threadIdx.x;
    __shared__ float imp[1024];
    if (i < N) {
        float cn = normCur[(size_t)b * N + i];
        float im;
        if (normPar) {
            int y = i / g, xx = i % g;
            int pcell = (2 * y) * (2 * g) + 2 * xx;          // nearest-interp pick
            im = fabsf(cn - normPar[(size_t)b * 4 * N + pcell]);
        } else {
            im = cn;
        }
        imp[i] = fminf(fmaxf(im, 1e-8f), 20000.0f);
    }
    __syncthreads();
    if (i < N) {
        float mi = imp[i];
        int rank = 0;
        for (int j = 0; j < N; ++j) {
            float mj = imp[j];
            rank += (mj > mi) || (mj == mi && j < i);        // jax top_k tie order
        }
        if (rank < k) {
            idxOut[b * NTOK + offTok + rank] = i;
            float inv = 1.0f / (float)g;
            int y = i / g, xx = i % g;
            float* co = coordsOut + ((size_t)b * NTOK + offTok + rank) * 4;
            co[0] = ((float)xx + 0.5f) * inv;
            co[1] = ((float)y  + 0.5f) * inv;
            co[2] = inv;
            co[3] = inv;
        }
    }
}

// ---- K5: gather kept features + LayerNorm, one wave32 per token ---------
__global__ void gather_ln_kernel(const float* __restrict__ ws, const int* __restrict__ idxBuf,
                                 const float* __restrict__ gamma, const float* __restrict__ beta,
                                 float* __restrict__ normedOut) {
    int wv   = (blockIdx.x * blockDim.x + threadIdx.x) >> 5;
    int lane = threadIdx.x & 31;
    if (wv >= B * NTOK) return;
    int b = wv / NTOK, pos = wv % NTOK;
    int l = 5;
    #pragma unroll
    for (int q = 0; q < 6; ++q)
        if (pos >= TOFF[q] && pos < TOFF[q + 1]) l = q;
    int N = NS[l];
    int cell = idxBuf[b * NTOK + pos];
    const float* p = ws + PYR_OFF[l] + (size_t)b * C * N + cell;
    float v[4], s = 0.f, sq = 0.f;
    #pragma unroll
    for (int i = 0; i < 4; ++i) {
        int c = lane + 32 * i;
        float t = p[(size_t)c * N];
        v[i] = t; s += t; sq = fmaf(t, t, sq);
    }
    #pragma unroll
    for (int m = 16; m >= 1; m >>= 1) {                       // wave32 reduction
        s  += __shfl_xor(s,  m, 32);
        sq += __shfl_xor(sq, m, 32);
    }
    float mu  = s * (1.0f / 128.0f);
    float var = sq * (1.0f / 128.0f) - mu * mu;
    float r   = rsqrtf(var + 1e-5f);
    float* o  = normedOut + (size_t)wv * C;
    #pragma unroll
    for (int i = 0; i < 4; ++i) {
        int c = lane + 32 * i;
        o[c] = (v[i] - mu) * r * gamma[c] + beta[c];
    }
}

// ---- K6: projection GEMM via V_WMMA_F32_16X16X4_F32 ---------------------
// One wave per 16(token) x 16(Cout) tile; K=128 in 32 steps of K=4.
// A 16x4 f32 layout: lanes 0-15 hold {K0,K1} of row M=lane; lanes 16-31 hold {K2,K3}.
// B 4x16 f32 layout mirrors it; C/D: VGPR v -> M=v (lanes 0-15) / M=v+8 (lanes 16-31), N=lane%16.
__global__ void gemm_wmma_kernel(const float* __restrict__ normed, const float* __restrict__ w,
                                 const float* __restrict__ bias, float* __restrict__ out) {
    int blk  = blockIdx.x;
    int nt   = blk & 7;                 // Cout tile
    int mt   = (blk >> 3) % 26;         // token tile (26*16 = 416 >= 408)
    int b    = blk / (26 * 8);
    int lane = threadIdx.x;
    int half = lane >> 4;               // 0: K pair {0,1}, 1: K pair {2,3}
    int l16  = lane & 15;

    int tokBase = mt * 16;
    int tokA = tokBase + l16;
    tokA = tokA > (NTOK - 1) ? (NTOK - 1) : tokA;            // branchless clamp
    const float* arow = normed + ((size_t)b * NTOK + tokA) * C + 2 * half;
    const float* brow = w + (size_t)(nt * 16 + l16) * C + 2 * half;

    v8f acc = {};
    #pragma unroll
    for (int kk = 0; kk < 32; ++kk) {
        v2f a  = *(const v2f*)(arow + 4 * kk);
        v2f bb = *(const v2f*)(brow + 4 * kk);
        acc = __builtin_amdgcn_wmma_f32_16x16x4_f32(
            /*neg_a=*/false, a, /*neg_b=*/false, bb,
            /*c_mod=*/(short)0, acc, /*reuse_a=*/false, /*reuse_b=*/false);
    }

    float bv = bias[nt * 16 + l16];
    #pragma unroll
    for (int v = 0; v < 8; ++v) {
        int tok = tokBase + v + 8 * half;
        if (tok < NTOK)
            out[((size_t)b * NTOK + tok) * COUT + nt * 16 + l16] = acc[v] + bv;
    }
}

__global__ void sparsity_kernel(float* __restrict__ p) {
    if (threadIdx.x == 0) *p = (float)NTOK / (float)(H * W);
}

extern "C" void kernel_launch(void* const* d_in, const int* in_sizes, int n_in,
                              void* d_out, int out_size, void* d_ws, size_t ws_size,
                              hipStream_t stream) {
    (void)in_sizes; (void)n_in; (void)out_size; (void)ws_size;
    const float* x     = (const float*)d_in[0];
    const float* gamma = (const float*)d_in[1];
    const float* beta  = (const float*)d_in[2];
    const float* w     = (const float*)d_in[3];
    const float* bias  = (const float*)d_in[4];
    float* out = (float*)d_out;
    float* ws  = (float*)d_ws;

    float* coords   = out + (size_t)B * NTOK * COUT;
    float* sparsity = coords + (size_t)B * NTOK * 4;
    int*   idxBuf   = (int*)(ws + IDX_OFF);
    float* normed   = ws + NORMED_OFF;

    // 1. streaming max-pool 512x512 -> 32x32 (the only BW-relevant pass, ~1 GiB, NT loads)
    pool512_kernel<<<B * C * 32, 256, 0, stream>>>(x, ws + PYR_OFF[5]);

    // 2. hierarchical 2x2 downsample for levels 4..0
    for (int l = 4; l >= 0; --l) {
        int total = B * C * NS[l];
        down_kernel<<<(total + 255) / 256, 256, 0, stream>>>(
            ws + PYR_OFF[l + 1], ws + PYR_OFF[l], 1 << l, NS[l]);
    }

    // 3. channel norms per level
    for (int l = 0; l < 6; ++l) {
        int total = B * NS[l];
        norm_kernel<<<(total + 255) / 256, 256, 0, stream>>>(
            ws + PYR_OFF[l], ws + NORM_OFF[l], NS[l]);
    }

    // 4. importance + top-k (exact jax order) + coords
    for (int l = 0; l < 6; ++l) {
        const float* npar = (l < 5) ? (ws + NORM_OFF[l + 1]) : nullptr;
        topk_kernel<<<B, 1024, 0, stream>>>(
            ws + NORM_OFF[l], npar, NS[l], 1 << l, KS[l], idxBuf, coords, TOFF[l]);
    }

    // 5. gather + LayerNorm (one wave32 per token)
    gather_ln_kernel<<<(B * NTOK) / 8, 256, 0, stream>>>(ws, idxBuf, gamma, beta, normed);

    // 6. fp32 WMMA projection [408x128] @ [128x128]^T + bias, per batch
    gemm_wmma_kernel<<<B * 26 * 8, 32, 0, stream>>>(normed, w, bias, out);

    // 7. scalar sparsity
    sparsity_kernel<<<1, 32, 0, stream>>>(sparsity);
}